// DGBlock_84971632984554
// MI455X (gfx1250) — compile-verified
//
#include <hip/hip_runtime.h>
#include <hip/hip_bf16.h>

#define NN 64
#define CC 64
#define TT 512
#define VV 18
#define KK 8
#define MID 8
#define S1 (TT*VV)      // 9216
#define S2 (TT*19)      // 9728

typedef float v2f __attribute__((ext_vector_type(2)));
typedef float v8f __attribute__((ext_vector_type(8)));

__device__ __forceinline__ v8f wmma_f32_k4(v2f a, v2f b, v8f c) {
  // V_WMMA_F32_16X16X4_F32 : D(16x16) = A(16x4) x B(4x16) + C
  return __builtin_amdgcn_wmma_f32_16x16x4_f32(false, a, false, b, (short)0, c, false, false);
}

// ---------------------------------------------------------------------------
// 1x1 conv as GEMM: out[n,o,s] = sum_c W[o,c]*in[n,c,s] + bias[o]
// Cout = Cin = 64. Block = 128 threads (4 waves), wave w -> m-tile w.
// grid.x = N * (S/16). S must be a multiple of 16 (9216 / 9728 both are).
// ---------------------------------------------------------------------------
__global__ void conv1x1_wmma(const float* __restrict__ in, const float* __restrict__ W,
                             const float* __restrict__ bias, float* __restrict__ out, int S)
{
  int numCT = S >> 4;
  int n  = blockIdx.x / numCT;
  int ct = blockIdx.x % numCT;
  int lane  = threadIdx.x & 31;
  int wave  = threadIdx.x >> 5;
  int lhalf = lane & 15;
  int hi    = lane >> 4;           // 0: K 0,1 ; 1: K 2,3
  int m0  = wave * 16;
  int col = ct * 16 + lhalf;
  const float* inN = in + (size_t)n * CC * S;
  v8f acc = {};
#pragma unroll
  for (int k = 0; k < CC; k += 4) {
    int kk = k + 2*hi;
    if (k + 8 < CC) __builtin_prefetch(&inN[(size_t)(kk+8)*S + col], 0, 3);
    v2f a, b;
    a.x = W[(m0 + lhalf)*CC + kk];
    a.y = W[(m0 + lhalf)*CC + kk + 1];
    b.x = inN[(size_t)kk     * S + col];
    b.y = inN[(size_t)(kk+1) * S + col];
    acc = wmma_f32_k4(a, b, acc);
  }
  float* outN = out + (size_t)n * CC * S;
  int mb = m0 + hi*8;              // C/D layout: lanes 0-15 rows r, 16-31 rows 8+r
#pragma unroll
  for (int r = 0; r < 8; ++r)
    outN[(size_t)(mb + r)*S + col] = acc[r] + bias[mb + r];
}

// Same GEMM but B-operand gets fused BN(scale,shift)+ReLU on load (tr conv).
__global__ void conv1x1_wmma_act(const float* __restrict__ in,
                                 const float* __restrict__ asc, const float* __restrict__ ash,
                                 const float* __restrict__ W, const float* __restrict__ bias,
                                 float* __restrict__ out, int S)
{
  int numCT = S >> 4;
  int n  = blockIdx.x / numCT;
  int ct = blockIdx.x % numCT;
  int lane = threadIdx.x & 31, wave = threadIdx.x >> 5;
  int lhalf = lane & 15, hi = lane >> 4;
  int m0 = wave * 16;
  int col = ct * 16 + lhalf;
  const float* inN = in + (size_t)n * CC * S;
  v8f acc = {};
#pragma unroll
  for (int k = 0; k < CC; k += 4) {
    int kk = k + 2*hi;
    v2f a, b;
    a.x = W[(m0 + lhalf)*CC + kk];
    a.y = W[(m0 + lhalf)*CC + kk + 1];
    b.x = fmaxf(inN[(size_t)kk     * S + col]*asc[kk]   + ash[kk],   0.f);
    b.y = fmaxf(inN[(size_t)(kk+1) * S + col]*asc[kk+1] + ash[kk+1], 0.f);
    acc = wmma_f32_k4(a, b, acc);
  }
  float* outN = out + (size_t)n * CC * S;
  int mb = m0 + hi*8;
#pragma unroll
  for (int r = 0; r < 8; ++r)
    outN[(size_t)(mb + r)*S + col] = acc[r] + bias[mb + r];
}

// ---------------------------------------------------------------------------
// Batched graph einsum: y[nc][t][w] = sum_v pre[nc][t][v] * A[nc][v][w]
// 4096 independent [512x18]x[18x18] GEMMs. grid.x = N*64*(T/16), block=64 (2 waves,
// wave -> w-tile of 16; tile1 covers cols 16..17 with masking).
// ---------------------------------------------------------------------------
__global__ void gcn_einsum_wmma(const float* __restrict__ pre, const float* __restrict__ Amat,
                                float* __restrict__ y)
{
  int Ttiles = TT / 16;
  int b = blockIdx.x;
  int tt = b % Ttiles; b /= Ttiles;
  int nc = b;                                  // n*64 + kc
  const float* P  = pre  + (size_t)nc * S1;
  const float* Am = Amat + (size_t)nc * VV * VV;
  int lane = threadIdx.x & 31, wave = threadIdx.x >> 5;
  int lhalf = lane & 15, hi = lane >> 4;
  int w0 = wave * 16, t0 = tt * 16;
  int wcol = w0 + lhalf;
  bool wOK = wcol < VV;
  v8f acc = {};
#pragma unroll
  for (int k = 0; k < 20; k += 4) {            // K = 18 padded to 20
    int ka = k + 2*hi;
    int trow = t0 + lhalf;
    v2f a, bb;
    a.x  = (ka     < VV) ? P[trow*VV + ka]     : 0.f;
    a.y  = (ka + 1 < VV) ? P[trow*VV + ka + 1] : 0.f;
    bb.x = (wOK && ka     < VV) ? Am[ka    *VV + wcol] : 0.f;
    bb.y = (wOK && ka + 1 < VV) ? Am[(ka+1)*VV + wcol] : 0.f;
    acc = wmma_f32_k4(a, bb, acc);
  }
  if (wOK) {
    float* Y = y + (size_t)nc * S1;
    int tb = t0 + hi*8;
#pragma unroll
    for (int r = 0; r < 8; ++r) Y[(tb + r)*VV + wcol] = acc[r];
  }
}

// ---------------------------------------------------------------------------
// BN statistics: one block per channel, deterministic tree reduction.
// src layout [N, 64, S]; writes fused affine scale/shift.
// ---------------------------------------------------------------------------
__global__ void bn_stats(const float* __restrict__ src, int S,
                         const float* __restrict__ g, const float* __restrict__ b,
                         float* __restrict__ scale, float* __restrict__ shift)
{
  int c = blockIdx.x;
  __shared__ float s1s[256], s2s[256];
  float s1 = 0.f, s2 = 0.f;
  size_t per = (size_t)NN * S;
  for (size_t i = threadIdx.x; i < per; i += blockDim.x) {
    size_t n = i / (size_t)S, s = i % (size_t)S;
    float v = src[(n*CC + c)*(size_t)S + s];
    s1 += v; s2 += v*v;
  }
  s1s[threadIdx.x] = s1; s2s[threadIdx.x] = s2; __syncthreads();
  for (int o = 128; o > 0; o >>= 1) {
    if ((int)threadIdx.x < o) { s1s[threadIdx.x] += s1s[threadIdx.x+o]; s2s[threadIdx.x] += s2s[threadIdx.x+o]; }
    __syncthreads();
  }
  if (threadIdx.x == 0) {
    float inv = 1.f / (float)per;
    float m   = s1s[0]*inv;
    float var = s2s[0]*inv - m*m;
    float rs  = rsqrtf(var + 1e-5f);
    scale[c] = g[c]*rs;
    shift[c] = b[c] - m*g[c]*rs;
  }
}

__global__ void bn_relu_kernel(const float* __restrict__ src, const float* __restrict__ sc,
                               const float* __restrict__ sh, float* __restrict__ dst, int S)
{
  size_t total = (size_t)NN*CC*S;
  for (size_t i = (size_t)blockIdx.x*blockDim.x + threadIdx.x; i < total;
       i += (size_t)gridDim.x*blockDim.x) {
    int c = (int)((i / (size_t)S) % CC);
    dst[i] = fmaxf(src[i]*sc[c] + sh[c], 0.f);
  }
}

__global__ void bn_addres_relu_kernel(const float* __restrict__ src, const float* __restrict__ sc,
                                      const float* __restrict__ sh, const float* __restrict__ res,
                                      float* __restrict__ dst, int S)
{
  size_t total = (size_t)NN*CC*S;
  for (size_t i = (size_t)blockIdx.x*blockDim.x + threadIdx.x; i < total;
       i += (size_t)gridDim.x*blockDim.x) {
    int c = (int)((i / (size_t)S) % CC);
    dst[i] = fmaxf(src[i]*sc[c] + sh[c] + res[i], 0.f);
  }
}

// tmp[n,c,v] = mean over T of x[n,c,t,v].  grid = N*C, block = 256 (252 used: 14 t-groups x 18 v)
__global__ void tmean_kernel(const float* __restrict__ x, float* __restrict__ tmp)
{
  __shared__ float p[14][VV];
  size_t nc = blockIdx.x;
  const float* X = x + nc * (size_t)S1;
  int gidx = threadIdx.x / VV, v = threadIdx.x % VV;
  if (threadIdx.x < 252) {
    float s = 0.f;
    for (int t = gidx; t < TT; t += 14) s += X[t*VV + v];
    p[gidx][v] = s;
  }
  __syncthreads();
  if (threadIdx.x < VV) {
    float s = 0.f;
    for (int gg = 0; gg < 14; ++gg) s += p[gg][threadIdx.x];
    tmp[nc*VV + threadIdx.x] = s * (1.f/TT);
  }
}

// x1/x2 = conv1x1(tmp) ; tiny GEMM per n
__global__ void x12_kernel(const float* __restrict__ tmp,
                           const float* __restrict__ w1, const float* __restrict__ b1,
                           const float* __restrict__ w2, const float* __restrict__ b2,
                           float* __restrict__ x1, float* __restrict__ x2)
{
  int n = blockIdx.x;
  const float* Tm = tmp + (size_t)n*CC*VV;
  for (int i = threadIdx.x; i < CC*VV; i += blockDim.x) {
    int o = i / VV, v = i % VV;
    float s1 = b1[o], s2 = b2[o];
    for (int c = 0; c < CC; ++c) {
      float tv = Tm[c*VV + v];
      s1 += w1[o*CC + c]*tv;
      s2 += w2[o*CC + c]*tv;
    }
    x1[(size_t)n*CC*VV + i] = s1;
    x2[(size_t)n*CC*VV + i] = s2;
  }
}

// A[n,k,c,v,w] = tanh(x1-x2)*alpha0 + softmax_v(x1^T x2)*beta0 + Ap[k]
// grid = N*K, block = 352 (324 active)
__global__ void amat_kernel(const float* __restrict__ x1, const float* __restrict__ x2,
                            const float* __restrict__ alpha, const float* __restrict__ beta,
                            const float* __restrict__ Ap, float* __restrict__ Am)
{
  int nk = blockIdx.x; int k = nk % KK; int n = nk / KK;
  __shared__ float X1[MID][VV], X2[MID][VV], Sm[VV][VV], mx[VV], sm[VV];
  int tid = threadIdx.x;
  if (tid < MID*VV) {
    int c = tid / VV, v = tid % VV;
    X1[c][v] = x1[((size_t)n*CC + k*MID + c)*VV + v];
    X2[c][v] = x2[((size_t)n*CC + k*MID + c)*VV + v];
  }
  __syncthreads();
  int v = tid / VV, w = tid % VV;
  if (tid < VV*VV) {
    float s = 0.f;
    for (int c = 0; c < MID; ++c) s += X1[c][v]*X2[c][w];
    Sm[v][w] = s;
  }
  __syncthreads();
  if (tid < VV) {                       // column softmax over v, w = tid
    float m = -1e30f;
    for (int vv = 0; vv < VV; ++vv) m = fmaxf(m, Sm[vv][tid]);
    float su = 0.f;
    for (int vv = 0; vv < VV; ++vv) su += expf(Sm[vv][tid] - m);
    mx[tid] = m; sm[tid] = su;
  }
  __syncthreads();
  if (tid < VV*VV) {
    float a0 = alpha[0], b0 = beta[0];
    float ada = expf(Sm[v][w] - mx[w]) / sm[w] * b0;
    float ap  = Ap[(size_t)k*VV*VV + v*VV + w];
    for (int c = 0; c < MID; ++c) {
      float tc = tanhf(X1[c][v] - X2[c][w]);
      Am[(((size_t)n*CC + k*MID + c)*VV + v)*VV + w] = tc*a0 + ada + ap;
    }
  }
}

// meanv[n,c,t] = mean over V.  grid = N*C, block = 256
__global__ void meanv_kernel(const float* __restrict__ g, float* __restrict__ mv)
{
  size_t nc = blockIdx.x;
  const float* G = g + nc * (size_t)S1;
  for (int t = threadIdx.x; t < TT; t += blockDim.x) {
    float s = 0.f;
    for (int v = 0; v < VV; ++v) s += G[t*VV + v];
    mv[nc*TT + t] = s * (1.f/VV);
  }
}

// xc[n,c,t,0..17] = gcn_out ; xc[n,c,t,18] = meanv
__global__ void xc_build(const float* __restrict__ g, const float* __restrict__ mv,
                         float* __restrict__ xc)
{
  size_t total = (size_t)NN*CC*S2;
  for (size_t i = (size_t)blockIdx.x*blockDim.x + threadIdx.x; i < total;
       i += (size_t)gridDim.x*blockDim.x) {
    int v = (int)(i % 19);
    size_t r = i / 19;
    int t = (int)(r % TT);
    size_t nc = r / TT;
    xc[i] = (v < VV) ? g[nc*S1 + (size_t)t*VV + v] : mv[nc*TT + t];
  }
}

// Pack 6 branch 1x1-conv weight/bias/bn params into one 64x64 GEMM
__global__ void pack_kernel(
  const float* w0, const float* w1, const float* w2, const float* w3, const float* w4, const float* w5,
  const float* b0, const float* b1, const float* b2, const float* b3, const float* b4, const float* b5,
  const float* g0, const float* g1, const float* g2, const float* g3, const float* g4,
  const float* e0, const float* e1, const float* e2, const float* e3, const float* e4,
  float* Wpk, float* bpk, float* gpk, float* epk)
{
  for (int i = threadIdx.x; i < CC*CC; i += blockDim.x) {
    int row = i / CC, c = i % CC;
    int b = (row < 14) ? 0 : (row < 24) ? 1 : (row < 34) ? 2 : (row < 44) ? 3 : (row < 54) ? 4 : 5;
    int base = (b == 0) ? 0 : 14 + (b-1)*10;
    const float* w = (b==0)?w0:(b==1)?w1:(b==2)?w2:(b==3)?w3:(b==4)?w4:w5;
    Wpk[i] = w[(row - base)*CC + c];
  }
  for (int row = threadIdx.x; row < CC; row += blockDim.x) {
    int b = (row < 14) ? 0 : (row < 24) ? 1 : (row < 34) ? 2 : (row < 44) ? 3 : (row < 54) ? 4 : 5;
    int base = (b == 0) ? 0 : 14 + (b-1)*10;
    const float* bb = (b==0)?b0:(b==1)?b1:(b==2)?b2:(b==3)?b3:(b==4)?b4:b5;
    bpk[row] = bb[row - base];
    if (b < 5) {
      const float* gg = (b==0)?g0:(b==1)?g1:(b==2)?g2:(b==3)?g3:g4;
      const float* ee = (b==0)?e0:(b==1)?e1:(b==2)?e2:(b==3)?e3:e4;
      gpk[row] = gg[row - base]; epk[row] = ee[row - base];
    } else {
      gpk[row] = 1.f; epk[row] = 0.f;
    }
  }
}

// Per-branch BN+ReLU then dilated 3x1 tconv (b0..b3) / maxpool3 (b4) / identity (b5).
// grid = N*T, block = 256, loop over 64*19 outputs
__global__ void branch_post(const float* __restrict__ raw, const float* __restrict__ sc,
                            const float* __restrict__ sh,
                            const float* __restrict__ tw0, const float* __restrict__ tb0,
                            const float* __restrict__ tw1, const float* __restrict__ tb1,
                            const float* __restrict__ tw2, const float* __restrict__ tb2,
                            const float* __restrict__ tw3, const float* __restrict__ tb3,
                            float* __restrict__ out)
{
  int nt = blockIdx.x;
  int t = nt % TT, n = nt / TT;
  const float* R = raw + (size_t)n*CC*S2;
  float*       O = out + (size_t)n*CC*S2;
  for (int idx = threadIdx.x; idx < CC*19; idx += blockDim.x) {
    int c = idx / 19, v = idx % 19;
    float o;
    if (c < 44) {
      int b    = (c < 14) ? 0 : 1 + (c - 14)/10;
      int base = (b == 0) ? 0 : 14 + (b-1)*10;
      int bc   = (b == 0) ? 14 : 10;
      int cl   = c - base;
      int dil  = b + 1;
      const float* tw = (b==0)?tw0:(b==1)?tw1:(b==2)?tw2:tw3;
      const float* tb = (b==0)?tb0:(b==1)?tb1:(b==2)?tb2:tb3;
      float acc = tb[cl];
      for (int j = 0; j < 3; ++j) {
        int ts = t + (j - 1)*dil;
        if (ts < 0 || ts >= TT) continue;
        for (int ci = 0; ci < bc; ++ci) {
          int ch = base + ci;
          float a = fmaxf(R[(size_t)ch*S2 + ts*19 + v]*sc[ch] + sh[ch], 0.f);
          acc += tw[(cl*bc + ci)*3 + j] * a;
        }
      }
      o = acc;
    } else if (c < 54) {
      float m = 0.f; bool first = true;
      for (int j = -1; j <= 1; ++j) {
        int ts = t + j; if (ts < 0 || ts >= TT) continue;
        float a = fmaxf(R[(size_t)c*S2 + ts*19 + v]*sc[c] + sh[c], 0.f);
        m = first ? a : fmaxf(m, a); first = false;
      }
      o = m;
    } else {
      o = R[(size_t)c*S2 + t*19 + v];
    }
    O[(size_t)c*S2 + t*19 + v] = o;
  }
}

// feat[n,c,t,v] = branch_out[..,v] + branch_out[..,18]*coef[v]  (v<18)
__global__ void feat_kernel(const float* __restrict__ bo, const float* __restrict__ coef,
                            float* __restrict__ out)
{
  size_t total = (size_t)NN*CC*TT*VV;
  for (size_t i = (size_t)blockIdx.x*blockDim.x + threadIdx.x; i < total;
       i += (size_t)gridDim.x*blockDim.x) {
    int v = (int)(i % VV);
    size_t r = i / VV;
    int t = (int)(r % TT);
    size_t nc = r / TT;
    const float* B = bo + nc*S2 + (size_t)t*19;
    out[i] = B[v] + B[18]*coef[v];
  }
}

// ---------------------------------------------------------------------------
extern "C" void kernel_launch(void* const* d_in, const int* in_sizes, int n_in,
                              void* d_out, int out_size, void* d_ws, size_t ws_size,
                              hipStream_t stream)
{
  (void)in_sizes; (void)n_in; (void)out_size; (void)ws_size;
  // Input order (setup_inputs insertion order):
  const float* x     = (const float*)d_in[0];
  const float* Agcn  = (const float*)d_in[1];
  const float* pre_w = (const float*)d_in[2];
  const float* pre_b = (const float*)d_in[3];
  const float* pre_g = (const float*)d_in[4];
  const float* pre_be= (const float*)d_in[5];
  const float* c1_w  = (const float*)d_in[6];
  const float* c1_b  = (const float*)d_in[7];
  const float* c2_w  = (const float*)d_in[8];
  const float* c2_b  = (const float*)d_in[9];
  const float* alpha = (const float*)d_in[10];
  const float* beta  = (const float*)d_in[11];
  const float* post_w= (const float*)d_in[12];
  const float* post_b= (const float*)d_in[13];
  const float* bn_g  = (const float*)d_in[14];
  const float* bn_b  = (const float*)d_in[15];
  // branches: 0:16-21, 1:22-27, 2:28-33, 3:34-39, 4:40-43, 5:44-45
  const float* bw[6]  = {(const float*)d_in[16], (const float*)d_in[22], (const float*)d_in[28],
                         (const float*)d_in[34], (const float*)d_in[40], (const float*)d_in[44]};
  const float* bbi[6] = {(const float*)d_in[17], (const float*)d_in[23], (const float*)d_in[29],
                         (const float*)d_in[35], (const float*)d_in[41], (const float*)d_in[45]};
  const float* bg[5]  = {(const float*)d_in[18], (const float*)d_in[24], (const float*)d_in[30],
                         (const float*)d_in[36], (const float*)d_in[42]};
  const float* bbe[5] = {(const float*)d_in[19], (const float*)d_in[25], (const float*)d_in[31],
                         (const float*)d_in[37], (const float*)d_in[43]};
  const float* tw[4]  = {(const float*)d_in[20], (const float*)d_in[26], (const float*)d_in[32],
                         (const float*)d_in[38]};
  const float* tb[4]  = {(const float*)d_in[21], (const float*)d_in[27], (const float*)d_in[33],
                         (const float*)d_in[39]};
  const float* add_coeff = (const float*)d_in[46];
  const float* tr_bng = (const float*)d_in[47];
  const float* tr_bnb = (const float*)d_in[48];
  const float* tr_w   = (const float*)d_in[49];
  const float* tr_b   = (const float*)d_in[50];
  const float* obn_g  = (const float*)d_in[51];
  const float* obn_b  = (const float*)d_in[52];

  // workspace layout (floats)
  const size_t REG = (size_t)NN*CC*S2;          // 39,845,888
  float* rA  = (float*)d_ws;
  float* rB  = rA + REG;
  float* rC  = rB + REG;
  float* tmp = rC + REG;                        // N*C*V
  float* x1  = tmp + (size_t)NN*CC*VV;
  float* x2  = x1  + (size_t)NN*CC*VV;
  float* Am  = x2  + (size_t)NN*CC*VV;          // N*64*V*V
  float* mv  = Am  + (size_t)NN*CC*VV*VV;       // N*64*T
  float* Wpk = mv  + (size_t)NN*CC*TT;
  float* bpk = Wpk + CC*CC;
  float* gpk = bpk + CC;
  float* epk = gpk + CC;
  float* st  = epk + CC;                        // 5 scale/shift pairs of 64
  float *sc0 = st+0*128,  *sh0 = st+0*128+64;   // pre bn
  float *sc1 = st+1*128,  *sh1 = st+1*128+64;   // post bn
  float *sc2 = st+2*128,  *sh2 = st+2*128+64;   // branch bn (packed)
  float *sc3 = st+3*128,  *sh3 = st+3*128+64;   // tr bn
  float *sc4 = st+4*128,  *sh4 = st+4*128+64;   // out bn

  const int EWG = 8192, EWB = 256;

  // ---- GCN ----
  conv1x1_wmma<<<dim3(NN*(S1/16)), dim3(128), 0, stream>>>(x, pre_w, pre_b, rA, S1);
  bn_stats<<<dim3(CC), dim3(256), 0, stream>>>(rA, S1, pre_g, pre_be, sc0, sh0);
  bn_relu_kernel<<<dim3(EWG), dim3(EWB), 0, stream>>>(rA, sc0, sh0, rB, S1);

  tmean_kernel<<<dim3(NN*CC), dim3(256), 0, stream>>>(x, tmp);
  x12_kernel<<<dim3(NN), dim3(256), 0, stream>>>(tmp, c1_w, c1_b, c2_w, c2_b, x1, x2);
  amat_kernel<<<dim3(NN*KK), dim3(352), 0, stream>>>(x1, x2, alpha, beta, Agcn, Am);

  gcn_einsum_wmma<<<dim3(NN*CC*(TT/16)), dim3(64), 0, stream>>>(rB, Am, rA);

  conv1x1_wmma<<<dim3(NN*(S1/16)), dim3(128), 0, stream>>>(rA, post_w, post_b, rC, S1);
  bn_stats<<<dim3(CC), dim3(256), 0, stream>>>(rC, S1, bn_g, bn_b, sc1, sh1);
  bn_addres_relu_kernel<<<dim3(EWG), dim3(EWB), 0, stream>>>(rC, sc1, sh1, x, rB, S1); // rB = gcn out

  // ---- TCN ----
  meanv_kernel<<<dim3(NN*CC), dim3(256), 0, stream>>>(rB, mv);
  xc_build<<<dim3(EWG), dim3(EWB), 0, stream>>>(rB, mv, rA);                           // rA = xc

  pack_kernel<<<dim3(1), dim3(256), 0, stream>>>(
      bw[0],bw[1],bw[2],bw[3],bw[4],bw[5],
      bbi[0],bbi[1],bbi[2],bbi[3],bbi[4],bbi[5],
      bg[0],bg[1],bg[2],bg[3],bg[4],
      bbe[0],bbe[1],bbe[2],bbe[3],bbe[4],
      Wpk, bpk, gpk, epk);

  conv1x1_wmma<<<dim3(NN*(S2/16)), dim3(128), 0, stream>>>(rA, Wpk, bpk, rC, S2);      // rC = branch conv raw
  bn_stats<<<dim3(CC), dim3(256), 0, stream>>>(rC, S2, gpk, epk, sc2, sh2);
  branch_post<<<dim3(NN*TT), dim3(256), 0, stream>>>(rC, sc2, sh2,
      tw[0],tb[0], tw[1],tb[1], tw[2],tb[2], tw[3],tb[3], rA);                          // rA = branch out

  feat_kernel<<<dim3(EWG), dim3(EWB), 0, stream>>>(rA, add_coeff, rB);                 // rB = feat
  bn_stats<<<dim3(CC), dim3(256), 0, stream>>>(rB, S1, tr_bng, tr_bnb, sc3, sh3);
  conv1x1_wmma_act<<<dim3(NN*(S1/16)), dim3(128), 0, stream>>>(rB, sc3, sh3, tr_w, tr_b, rC, S1);
  bn_stats<<<dim3(CC), dim3(256), 0, stream>>>(rC, S1, obn_g, obn_b, sc4, sh4);
  bn_addres_relu_kernel<<<dim3(EWG), dim3(EWB), 0, stream>>>(rC, sc4, sh4, x, (float*)d_out, S1);
}